// BCRBlock_24111946400594
// MI455X (gfx1250) — compile-verified
//
#include <hip/hip_runtime.h>
#include <math.h>

// ---------------------------------------------------------------------------
// BCRBlock forward for MI455X (gfx1250, wave32).
// fp32 end-to-end; all GEMM-like work on V_WMMA_F32_16X16X4_F32.
//
// Input pointer order = setup_inputs() dict insertion order:
//   0:x 1:residual
//   rn1: 2:w1 3:b1 4:g1 5:be1 6:w2 7:b2 8:g2 9:be2 10:rw 11:rb
//   be : 12:pw 13:pb 14:pg 15:pbe 16:kw 17:kb 18:kg 19:kbe
//        20:lw1 21:lb1 22:lg1 23:lbe1 24:lw2 25:lb2 26:lg2 27:lbe2
//        28:gw1 29:gb1 30:gg1 31:gbe1 32:gw2 33:gb2 34:gg2 35:gbe2
//   rn2[i]: 36+8i .. 43+8i : w1,b1,g1,be1,w2,b2,g2,be2
// ---------------------------------------------------------------------------

#define Bb   16
#define Cc   128
#define Nn   2000
#define KNN  9
#define EPSf 1e-5f
#define TOT  (Bb * Cc * Nn)

typedef float v2f __attribute__((ext_vector_type(2)));
typedef float v8f __attribute__((ext_vector_type(8)));

// ---------------- elementwise add: t0 = x + residual -----------------------
__global__ __launch_bounds__(256) void k_add(const float* __restrict__ a,
                                             const float* __restrict__ b,
                                             float* __restrict__ y) {
  int i = blockIdx.x * 256 + threadIdx.x;
  if (i < TOT) y[i] = a[i] + b[i];
}

// ---------------- 1x1 conv (GEMM W[128x128] @ X[b][128xN]) via WMMA --------
// Block = 8 waves; wave w computes o-tile w (16 rows) for one 16-col strip.
// X strip staged once in LDS, paired (k,k+1) so B-frag is one ds_load_b64.
// WMMA fragment layout per CDNA5 ISA 7.12.2:
//   A: lane l<16 -> M=l, vgpr0=K+0, vgpr1=K+1 ; lanes 16-31 -> K+2,K+3
//   B: mirrored with N on lanes
//   D: vgpr r, lanes 0-15 -> row r ; lanes 16-31 -> row 8+r ; col = lane&15
__global__ __launch_bounds__(256)
void k_conv1x1(const float* __restrict__ W, const float* __restrict__ bias,
               const float* __restrict__ X, float* __restrict__ Y) {
  __shared__ float lds[Cc * 16];  // element (k,n) at ((k>>1)*16+n)*2 + (k&1)
  const int tid = threadIdx.x;
  const int wave = tid >> 5, lane = tid & 31;
  const int half = lane >> 4, l = lane & 15;
  const int n0 = blockIdx.x * 16, b = blockIdx.z;
  const int o0 = wave * 16;
  const float* Xb = X + (size_t)b * Cc * Nn;
#pragma unroll
  for (int i = 0; i < (Cc * 16) / 256; ++i) {  // 8
    int id = tid + i * 256;
    int k = id >> 4, n = id & 15;
    lds[((k >> 1) * 16 + n) * 2 + (k & 1)] = Xb[k * Nn + n0 + n];
  }
  __syncthreads();
  v8f acc = {};
  const v2f* lds2 = (const v2f*)lds;
  const float* Wrow = W + (o0 + l) * Cc;
#pragma unroll
  for (int k = 0; k < Cc; k += 4) {
    const int ka = k + half * 2;
    v2f a = *(const v2f*)&Wrow[ka];        // global_load_b64
    v2f bf = lds2[(ka >> 1) * 16 + l];     // ds_load_b64
    acc = __builtin_amdgcn_wmma_f32_16x16x4_f32(false, a, false, bf,
                                                (short)0, acc, false, false);
  }
  float* Yb = Y + (size_t)b * Cc * Nn;
#pragma unroll
  for (int r = 0; r < 8; ++r) {
    int o = o0 + half * 8 + r;
    Yb[o * Nn + n0 + l] = acc[r] + bias[o];
  }
}

// ---------------- per-(b,c) mean/var over `inner` --------------------------
__global__ __launch_bounds__(256)
void k_stats_bc(const float* __restrict__ X, float* __restrict__ m,
                float* __restrict__ v, int inner) {
  __shared__ float rs[256], rq[256];
  const int bc = blockIdx.x;
  const float* p = X + (size_t)bc * inner;
  float s = 0.f, q = 0.f;
  for (int i = threadIdx.x; i < inner; i += 256) {
    float x = p[i];
    s += x; q += x * x;
  }
  rs[threadIdx.x] = s; rq[threadIdx.x] = q;
  __syncthreads();
  for (int off = 128; off > 0; off >>= 1) {
    if (threadIdx.x < off) {
      rs[threadIdx.x] += rs[threadIdx.x + off];
      rq[threadIdx.x] += rq[threadIdx.x + off];
    }
    __syncthreads();
  }
  if (threadIdx.x == 0) {
    float mean = rs[0] / (float)inner;
    m[bc] = mean;
    v[bc] = rq[0] / (float)inner - mean * mean;
  }
}

// ---------------- per-channel mean/var over (b, inner) ---------------------
__global__ __launch_bounds__(256)
void k_stats_c(const float* __restrict__ X, float* __restrict__ m,
               float* __restrict__ v, int inner) {
  __shared__ float rs[256], rq[256];
  const int c = blockIdx.x;
  float s = 0.f, q = 0.f;
  for (int b = 0; b < Bb; ++b) {
    const float* p = X + ((size_t)b * Cc + c) * inner;
    for (int i = threadIdx.x; i < inner; i += 256) {
      float x = p[i];
      s += x; q += x * x;
    }
  }
  rs[threadIdx.x] = s; rq[threadIdx.x] = q;
  __syncthreads();
  for (int off = 128; off > 0; off >>= 1) {
    if (threadIdx.x < off) {
      rs[threadIdx.x] += rs[threadIdx.x + off];
      rq[threadIdx.x] += rq[threadIdx.x + off];
    }
    __syncthreads();
  }
  if (threadIdx.x == 0) {
    float denom = (float)(Bb) * (float)inner;
    float mean = rs[0] / denom;
    m[c] = mean;
    v[c] = rq[0] / denom - mean * mean;
  }
}

// per-channel BN variance of instance-normalized data: mean_b v/(v+eps)
__global__ void k_chan(const float* __restrict__ v_bc, float* __restrict__ vbn) {
  int c = threadIdx.x;
  float s = 0.f;
  for (int b = 0; b < Bb; ++b) {
    float v = v_bc[b * Cc + c];
    s += v / (v + EPSf);
  }
  vbn[c] = s / (float)Bb;
}

// y = relu( g*( (x-m_bc)*rsqrt(v_bc+eps) )*rsqrt(vbn+eps) + be  [+ add] )
__global__ __launch_bounds__(256)
void k_apply_innorm(const float* __restrict__ X, const float* __restrict__ add,
                    float* __restrict__ Y, const float* __restrict__ m_bc,
                    const float* __restrict__ v_bc, const float* __restrict__ vbn,
                    const float* __restrict__ g, const float* __restrict__ be,
                    int addmode) {
  int i = blockIdx.x * 256 + threadIdx.x;
  if (i >= TOT) return;
  int bc = i / Nn;
  int c = bc % Cc;
  float y = (X[i] - m_bc[bc]) * rsqrtf(v_bc[bc] + EPSf) * rsqrtf(vbn[c] + EPSf);
  y = g[c] * y + be[c];
  if (addmode) y += add[i];
  Y[i] = fmaxf(y, 0.f);
}

// plain BN apply (stats over batch): y = g*(x-m)*rsqrt(v+eps)+be [, relu]
__global__ __launch_bounds__(256)
void k_apply_bn(const float* __restrict__ X, float* __restrict__ Y,
                const float* __restrict__ m_c, const float* __restrict__ v_c,
                const float* __restrict__ g, const float* __restrict__ be,
                int inner, int total, int do_relu) {
  int i = blockIdx.x * 256 + threadIdx.x;
  if (i >= total) return;
  int c = (i / inner) % Cc;
  float y = g[c] * (X[i] - m_c[c]) * rsqrtf(v_c[c] + EPSf) + be[c];
  Y[i] = do_relu ? fmaxf(y, 0.f) : y;
}

// ---------------- squared norms per point ----------------------------------
__global__ __launch_bounds__(256)
void k_sq(const float* __restrict__ X, float* __restrict__ sq) {
  int i = blockIdx.x * 256 + threadIdx.x;
  if (i >= Bb * Nn) return;
  int b = i / Nn, n = i % Nn;
  const float* p = X + (size_t)b * Cc * Nn + n;
  float s = 0.f;
  for (int c = 0; c < Cc; ++c) {
    float x = p[c * Nn];
    s += x * x;
  }
  sq[i] = s;
}

// ---------------- fused Gram(WMMA) + stable top-9 kNN ----------------------
// Block = 8 waves. Row strip A staged once in LDS (paired layout). Each wave
// computes one 16x16 Gram tile per chunk (pd into LDS); 16 scanner threads
// (tid%16==0, spread over all waves) merge 128 cands/chunk into sorted top-9.
__global__ __launch_bounds__(256)
void k_knn(const float* __restrict__ X, const float* __restrict__ sq,
           int* __restrict__ idx) {
  __shared__ float ldsA[Cc * 16];
  __shared__ float pd[16][132];
  __shared__ float tv[16][KNN];
  __shared__ int ti[16][KNN];
  const int tid = threadIdx.x;
  const int wave = tid >> 5, lane = tid & 31;
  const int half = lane >> 4, l = lane & 15;
  const int rt = blockIdx.x, b = blockIdx.z;
  const int n0 = rt * 16;
  const float* Xb = X + (size_t)b * Cc * Nn;
  const float* sqb = sq + b * Nn;

#pragma unroll
  for (int i = 0; i < (Cc * 16) / 256; ++i) {  // 8
    int id = tid + i * 256;
    int k = id >> 4, r = id & 15;
    ldsA[((k >> 1) * 16 + r) * 2 + (k & 1)] = Xb[k * Nn + n0 + r];
  }
  if (tid < 16 * KNN) {
    tv[tid / KNN][tid % KNN] = -3.402823466e38f;
    ti[tid / KNN][tid % KNN] = 0;
  }
  __syncthreads();

  const v2f* A2 = (const v2f*)ldsA;
  const int NTILE = Nn / 16;           // 125
  const int NCHUNK = (NTILE + 7) / 8;  // 16
  for (int ch = 0; ch < NCHUNK; ++ch) {
    int ct = ch * 8 + wave;
    if (ct < NTILE) {
      int m0 = ct * 16;
      v8f acc = {};
#pragma unroll 8
      for (int k = 0; k < Cc; k += 4) {
        int ka = k + half * 2;
        v2f a = A2[(ka >> 1) * 16 + l];
        v2f bf;
        bf.x = Xb[ka * Nn + m0 + l];
        bf.y = Xb[(ka + 1) * Nn + m0 + l];
        acc = __builtin_amdgcn_wmma_f32_16x16x4_f32(false, a, false, bf,
                                                    (short)0, acc, false, false);
      }
#pragma unroll
      for (int r = 0; r < 8; ++r) {
        int row = half * 8 + r;
        pd[row][wave * 16 + l] = 2.f * acc[r] - sqb[n0 + row] - sqb[m0 + l];
      }
    }
    __syncthreads();
    if ((tid & 15) == 0) {
      int row = tid >> 4;
      int base = ch * 128;
      int ncol = Nn - base;
      if (ncol > 128) ncol = 128;
      for (int j = 0; j < ncol; ++j) {
        float v = pd[row][j];
        if (v > tv[row][KNN - 1]) {
          int p = KNN - 1;
          while (p > 0 && v > tv[row][p - 1]) {
            tv[row][p] = tv[row][p - 1];
            ti[row][p] = ti[row][p - 1];
            --p;
          }
          tv[row][p] = v;
          ti[row][p] = base + j;
        }
      }
    }
    __syncthreads();
  }
  if ((tid & 15) == 0) {
    int row = tid >> 4;
    for (int k = 0; k < KNN; ++k)
      idx[((b * Nn) + n0 + row) * KNN + k] = ti[row][k];
  }
}

// ---------------- weight prep for edge convs -------------------------------
// WS[o][c]    = sum_t ( lw1[o,c,t] + lw1[o,128+c,t] )   (xe-part 1x1 weights)
// WnNeg[o][t*128+c] = -lw1[o,128+c,t]                   (neighbor GEMM, negated)
// W2t  [o][t*128+c] =  lw2[o,c,t]                       (lconv2 GEMM weights)
__global__ __launch_bounds__(256)
void k_prepw(const float* __restrict__ lw1, const float* __restrict__ lw2,
             float* __restrict__ WS, float* __restrict__ WnNeg,
             float* __restrict__ W2t) {
  int id = blockIdx.x * 256 + threadIdx.x;
  if (id >= Cc * Cc) return;
  int o = id / Cc, c = id % Cc;
  const float* wo = lw1 + o * (2 * Cc * 3);
  float s = 0.f;
  for (int t = 0; t < 3; ++t) {
    float we = wo[c * 3 + t], wn = wo[(Cc + c) * 3 + t];
    s += we + wn;
    WnNeg[o * 384 + t * Cc + c] = -wn;
    W2t[o * 384 + t * Cc + c] = lw2[o * 384 + c * 3 + t];
  }
  WS[id] = s;
}

// ---------------- edge conv 1 neighbor GEMM (WMMA, K=384, gathered B) ------
// out[b,o,col] = xe[b,o,n(col)] + sum_k WnNeg[o,k] * xl[c(k), idx[n,3j+t(k)]]
// cols = n*3+j, 6000 per batch = 375 tiles of 16. K order: k = t*128+c.
__global__ __launch_bounds__(256)
void k_lconv1_wmma(const float* __restrict__ X, const int* __restrict__ idx,
                   const float* __restrict__ Wn, const float* __restrict__ xe,
                   float* __restrict__ Y) {
  __shared__ float ldsV[384 * 16];  // element (k,col) at ((k>>1)*16+col)*2+(k&1)
  __shared__ int sidx[48];
  const int tid = threadIdx.x;
  const int wave = tid >> 5, lane = tid & 31;
  const int half = lane >> 4, l = lane & 15;
  const int col0 = blockIdx.x * 16, b = blockIdx.z;
  const int o0 = wave * 16;
  const float* Xb = X + (size_t)b * Cc * Nn;
  if (tid < 48) {
    int col = col0 + tid / 3, t = tid % 3;
    int n = col / 3, j = col % 3;
    sidx[tid] = idx[((size_t)b * Nn + n) * KNN + 3 * j + t];
  }
  __syncthreads();
#pragma unroll
  for (int i = 0; i < (384 * 16) / 256; ++i) {  // 24
    int id = tid + i * 256;
    int k = id >> 4, col = id & 15;
    int t = k >> 7, c = k & 127;
    int m = sidx[col * 3 + t];
    ldsV[((k >> 1) * 16 + col) * 2 + (k & 1)] = Xb[c * Nn + m];
  }
  __syncthreads();
  v8f acc = {};
  const v2f* V2 = (const v2f*)ldsV;
  const float* Wrow = Wn + (o0 + l) * 384;
#pragma unroll 8
  for (int k = 0; k < 384; k += 4) {
    int ka = k + half * 2;
    v2f a = *(const v2f*)&Wrow[ka];
    v2f bf = V2[(ka >> 1) * 16 + l];
    acc = __builtin_amdgcn_wmma_f32_16x16x4_f32(false, a, false, bf,
                                                (short)0, acc, false, false);
  }
  const int cl = col0 + l;
  const int n = cl / 3, j = cl % 3;
#pragma unroll
  for (int r = 0; r < 8; ++r) {
    int o = o0 + half * 8 + r;
    Y[(((size_t)b * Cc + o) * Nn + n) * 3 + j] =
        acc[r] + xe[((size_t)b * Cc + o) * Nn + n];
  }
}

// ---------------- edge conv 2 GEMM (WMMA, K=384) ---------------------------
// out[b,o,n] = lb2[o] + sum_k W2t[o,k] * H[b, c(k), n, t(k)]
__global__ __launch_bounds__(256)
void k_lconv2_wmma(const float* __restrict__ H, const float* __restrict__ W2t,
                   const float* __restrict__ bias, float* __restrict__ Y) {
  __shared__ float ldsV[384 * 16];
  const int tid = threadIdx.x;
  const int wave = tid >> 5, lane = tid & 31;
  const int half = lane >> 4, l = lane & 15;
  const int n0 = blockIdx.x * 16, b = blockIdx.z;
  const int o0 = wave * 16;
  const float* Hb = H + (size_t)b * Cc * Nn * 3;
#pragma unroll
  for (int i = 0; i < (384 * 16) / 256; ++i) {  // 24
    int id = tid + i * 256;
    int k = id >> 4, col = id & 15;
    int t = k >> 7, c = k & 127;
    ldsV[((k >> 1) * 16 + col) * 2 + (k & 1)] =
        Hb[((size_t)c * Nn + n0 + col) * 3 + t];
  }
  __syncthreads();
  v8f acc = {};
  const v2f* V2 = (const v2f*)ldsV;
  const float* Wrow = W2t + (o0 + l) * 384;
#pragma unroll 8
  for (int k = 0; k < 384; k += 4) {
    int ka = k + half * 2;
    v2f a = *(const v2f*)&Wrow[ka];
    v2f bf = V2[(ka >> 1) * 16 + l];
    acc = __builtin_amdgcn_wmma_f32_16x16x4_f32(false, a, false, bf,
                                                (short)0, acc, false, false);
  }
#pragma unroll
  for (int r = 0; r < 8; ++r) {
    int o = o0 + half * 8 + r;
    Y[((size_t)b * Cc + o) * Nn + n0 + l] = acc[r] + bias[o];
  }
}

// ---------------- tiny SE-style global branch (single block) ---------------
__global__ __launch_bounds__(512)
void k_gbranch(const float* __restrict__ gpool, const float* __restrict__ gw1,
               const float* __restrict__ gb1, const float* __restrict__ gg1,
               const float* __restrict__ gbe1, const float* __restrict__ gw2,
               const float* __restrict__ gb2, const float* __restrict__ gg2,
               const float* __restrict__ gbe2, float* __restrict__ gvec) {
  const int IC = Cc / 4;  // 32
  __shared__ float s1[Bb * 32], sm1[32], sv1[32];
  __shared__ float s2[Bb * Cc], sm2[Cc], sv2[Cc];
  const int tid = threadIdx.x;
  {
    int b = tid / IC, i = tid % IC;
    float acc = gb1[i];
    for (int c = 0; c < Cc; ++c) acc += gw1[i * Cc + c] * gpool[b * Cc + c];
    s1[tid] = acc;
  }
  __syncthreads();
  if (tid < IC) {
    float s = 0.f, q = 0.f;
    for (int b = 0; b < Bb; ++b) {
      float x = s1[b * IC + tid];
      s += x; q += x * x;
    }
    float mean = s / (float)Bb;
    sm1[tid] = mean;
    sv1[tid] = q / (float)Bb - mean * mean;
  }
  __syncthreads();
  {
    int i = tid % IC;
    float y = gg1[i] * (s1[tid] - sm1[i]) * rsqrtf(sv1[i] + EPSf) + gbe1[i];
    s1[tid] = fmaxf(y, 0.f);
  }
  __syncthreads();
  for (int u = 0; u < (Bb * Cc) / 512; ++u) {
    int t = tid + u * 512;
    int b = t / Cc, o = t % Cc;
    float acc = gb2[o];
    for (int i = 0; i < IC; ++i) acc += gw2[o * IC + i] * s1[b * IC + i];
    s2[t] = acc;
  }
  __syncthreads();
  if (tid < Cc) {
    float s = 0.f, q = 0.f;
    for (int b = 0; b < Bb; ++b) {
      float x = s2[b * Cc + tid];
      s += x; q += x * x;
    }
    float mean = s / (float)Bb;
    sm2[tid] = mean;
    sv2[tid] = q / (float)Bb - mean * mean;
  }
  __syncthreads();
  for (int u = 0; u < (Bb * Cc) / 512; ++u) {
    int t = tid + u * 512;
    int o = t % Cc;
    gvec[t] = gg2[o] * (s2[t] - sm2[o]) * rsqrtf(sv2[o] + EPSf) + gbe2[o];
  }
}

// ---------------- wei = sigmoid(g + local); h = 2x*wei + 2res*(1-wei) ------
__global__ __launch_bounds__(256)
void k_wei_mix(const float* __restrict__ x, const float* __restrict__ res,
               const float* __restrict__ local, const float* __restrict__ gvec,
               float* __restrict__ out) {
  int i = blockIdx.x * 256 + threadIdx.x;
  if (i >= TOT) return;
  int bc = i / Nn;
  int b = bc / Cc, c = bc % Cc;
  float z = gvec[b * Cc + c] + local[i];
  float w = 1.f / (1.f + expf(-z));
  out[i] = 2.f * x[i] * w + 2.f * res[i] * (1.f - w);
}

// ===========================================================================
extern "C" void kernel_launch(void* const* d_in, const int* in_sizes, int n_in,
                              void* d_out, int out_size, void* d_ws,
                              size_t ws_size, hipStream_t stream) {
  const float* x = (const float*)d_in[0];
  const float* res = (const float*)d_in[1];
  const float* r1_w1 = (const float*)d_in[2];
  const float* r1_b1 = (const float*)d_in[3];
  const float* r1_g1 = (const float*)d_in[4];
  const float* r1_be1 = (const float*)d_in[5];
  const float* r1_w2 = (const float*)d_in[6];
  const float* r1_b2 = (const float*)d_in[7];
  const float* r1_g2 = (const float*)d_in[8];
  const float* r1_be2 = (const float*)d_in[9];
  const float* r1_rw = (const float*)d_in[10];
  const float* r1_rb = (const float*)d_in[11];
  const float* pw = (const float*)d_in[12];
  const float* pb = (const float*)d_in[13];
  const float* pg = (const float*)d_in[14];
  const float* pbe = (const float*)d_in[15];
  const float* kw = (const float*)d_in[16];
  const float* kb = (const float*)d_in[17];
  const float* kg = (const float*)d_in[18];
  const float* kbe = (const float*)d_in[19];
  const float* lw1 = (const float*)d_in[20];
  const float* lb1 = (const float*)d_in[21];
  const float* lg1 = (const float*)d_in[22];
  const float* lbe1 = (const float*)d_in[23];
  const float* lw2 = (const float*)d_in[24];
  const float* lb2 = (const float*)d_in[25];
  const float* lg2 = (const float*)d_in[26];
  const float* lbe2 = (const float*)d_in[27];
  const float* gw1 = (const float*)d_in[28];
  const float* gb1 = (const float*)d_in[29];
  const float* gg1 = (const float*)d_in[30];
  const float* gbe1 = (const float*)d_in[31];
  const float* gw2 = (const float*)d_in[32];
  const float* gb2 = (const float*)d_in[33];
  const float* gg2 = (const float*)d_in[34];
  const float* gbe2 = (const float*)d_in[35];

  // workspace layout (floats); needs ~117 MB
  float* ws = (float*)d_ws;
  const size_t T = (size_t)TOT;
  float* buf0 = ws;          // t0 -> xe-part -> local/wei -> rn2 h2
  float* buf1 = ws + T;      // rn1 shortcut -> be x1 -> mix output h
  float* buf2 = ws + 2 * T;  // rn1 h1 -> xl
  float* buf3 = ws + 3 * T;  // rn1 out t1 -> rn2 h1
  float* bufH = ws + 4 * T;  // 3T : lconv1 output [B,128,N,3]
  float* sm = ws + 7 * T;
  float* m_bc = sm;              // 2048
  float* v_bc = sm + 2048;       // 2048
  float* vbn = sm + 4096;        // 128
  float* m_c = sm + 4224;        // 128
  float* v_c = sm + 4352;        // 128
  float* sqb = sm + 4480;        // 32000
  float* gpool = sm + 36480;     // 2048
  float* gvar = sm + 38528;      // 2048
  float* gvec = sm + 40576;      // 2048
  int* idx = (int*)(sm + 42624); // 288000 ints
  float* WS = sm + 330624;       // 16384
  float* WnNeg = sm + 347008;    // 49152
  float* W2t = sm + 396160;      // 49152

  const dim3 gConv(Nn / 16, 1, Bb);  // 125 x 1 x 16, 256 threads (8 waves)
  const int gElem = (TOT + 255) / 256;

  auto conv = [&](const float* W, const float* bs, const float* X, float* Y) {
    k_conv1x1<<<gConv, 256, 0, stream>>>(W, bs, X, Y);
  };
  auto innorm_bn = [&](float* X, const float* add, float* Y, const float* g,
                       const float* be, int addmode) {
    k_stats_bc<<<Bb * Cc, 256, 0, stream>>>(X, m_bc, v_bc, Nn);
    k_chan<<<1, Cc, 0, stream>>>(v_bc, vbn);
    k_apply_innorm<<<gElem, 256, 0, stream>>>(X, add, Y, m_bc, v_bc, vbn, g, be,
                                              addmode);
  };
  auto bn_c = [&](float* X, float* Y, const float* g, const float* be,
                  int inner, int total, int relu) {
    k_stats_c<<<Cc, 256, 0, stream>>>(X, m_c, v_c, inner);
    k_apply_bn<<<(total + 255) / 256, 256, 0, stream>>>(X, Y, m_c, v_c, g, be,
                                                        inner, total, relu);
  };

  // independent weight prep + t0 = x + residual
  k_prepw<<<(Cc * Cc + 255) / 256, 256, 0, stream>>>(lw1, lw2, WS, WnNeg, W2t);
  k_add<<<gElem, 256, 0, stream>>>(x, res, buf0);

  // ---- rn1 (pre=True) ----
  conv(r1_rw, r1_rb, buf0, buf1);  // shortcut
  conv(r1_w1, r1_b1, buf0, buf2);
  innorm_bn(buf2, nullptr, buf2, r1_g1, r1_be1, 0);
  conv(r1_w2, r1_b2, buf2, buf3);
  innorm_bn(buf3, buf1, buf3, r1_g2, r1_be2, 1);  // t1 in buf3

  // ---- be ----
  conv(pw, pb, buf3, buf1);
  bn_c(buf1, buf1, pg, pbe, Nn, TOT, 1);  // x1 in buf1
  k_stats_bc<<<Bb * Cc, 256, 0, stream>>>(buf1, gpool, gvar, Nn);  // avg pool
  conv(kw, kb, buf1, buf2);
  bn_c(buf2, buf2, kg, kbe, Nn, TOT, 1);  // xl in buf2
  k_sq<<<(Bb * Nn + 255) / 256, 256, 0, stream>>>(buf2, sqb);
  k_knn<<<dim3(Nn / 16, 1, Bb), 256, 0, stream>>>(buf2, sqb, idx);
  conv(WS, lb1, buf2, buf0);  // xe-part of edge conv (incl. bias lb1) -> buf0
  k_lconv1_wmma<<<dim3((Nn * 3) / 16, 1, Bb), 256, 0, stream>>>(buf2, idx,
                                                                WnNeg, buf0,
                                                                bufH);
  bn_c(bufH, bufH, lg1, lbe1, Nn * 3, TOT * 3, 1);
  k_lconv2_wmma<<<gConv, 256, 0, stream>>>(bufH, W2t, lb2, buf0);
  bn_c(buf0, buf0, lg2, lbe2, Nn, TOT, 1);  // local in buf0
  k_gbranch<<<1, 512, 0, stream>>>(gpool, gw1, gb1, gg1, gbe1, gw2, gb2, gg2,
                                   gbe2, gvec);
  k_wei_mix<<<gElem, 256, 0, stream>>>(x, res, buf0, gvec, buf1);  // h in buf1

  // ---- rn2 x3 (pre=False) ----
  for (int it = 0; it < 3; ++it) {
    const float* w1 = (const float*)d_in[36 + it * 8 + 0];
    const float* b1 = (const float*)d_in[36 + it * 8 + 1];
    const float* g1 = (const float*)d_in[36 + it * 8 + 2];
    const float* be1 = (const float*)d_in[36 + it * 8 + 3];
    const float* w2 = (const float*)d_in[36 + it * 8 + 4];
    const float* b2 = (const float*)d_in[36 + it * 8 + 5];
    const float* g2 = (const float*)d_in[36 + it * 8 + 6];
    const float* be2 = (const float*)d_in[36 + it * 8 + 7];
    conv(w1, b1, buf1, buf3);
    innorm_bn(buf3, nullptr, buf3, g1, be1, 0);
    conv(w2, b2, buf3, buf0);
    float* dst = (it == 2) ? (float*)d_out : buf1;
    innorm_bn(buf0, buf1, dst, g2, be2, 1);
  }
}